// LIMADNN2_42013370090068
// MI455X (gfx1250) — compile-verified
//
#include <hip/hip_runtime.h>

typedef __attribute__((ext_vector_type(2))) float v2f;
typedef __attribute__((ext_vector_type(8))) float v8f;

#define ROW_STRIDE 780   // NB * F = 65 * 12 floats per batch row

static __device__ __forceinline__ v8f wmma4(v2f a, v2f b, v8f c) {
  // V_WMMA_F32_16X16X4_F32: D = A(16x4) * B(4x16) + C(16x16), fp32 throughout
  return __builtin_amdgcn_wmma_f32_16x16x4_f32(
      /*neg_a=*/false, a, /*neg_b=*/false, b,
      /*c_mod=*/(short)0, c, /*reuse_a=*/false, /*reuse_b=*/false);
}

__global__ __launch_bounds__(256) void LIMADNN2_mlp_wmma(
    const float* __restrict__ x,
    const float* __restrict__ W1, const float* __restrict__ b1,
    const float* __restrict__ W2, const float* __restrict__ b2,
    const float* __restrict__ Ws, const float* __restrict__ bs,
    const float* __restrict__ Wo, const float* __restrict__ bo,
    float* __restrict__ out, float* __restrict__ fprev, int B)
{
  __shared__ float lds[8][256];            // one 16x16 f32 tile per wave

  const int lane  = threadIdx.x & 31;
  const int wave  = threadIdx.x >> 5;
  const int tile  = blockIdx.x * 8 + wave;
  const int r0    = tile * 16;
  if (r0 >= B) return;                     // wave-uniform exit; EXEC stays all-1s

  const int m  = lane & 15;                // row (A/C) or column N (B/C)
  const int kh = (lane >> 4) * 2;          // K half-offset within a K=4 chunk
  const int ch = (lane >> 4) * 8;          // C-layout row half
  float* t = &lds[wave][0];

  // ---- A fragments, layer 1: q = x[row, 0, 0:12], K chunks {0-3,4-7,8-11}
  const float* qrow = x + (size_t)(r0 + m) * ROW_STRIDE;
  v2f a1[3];
  #pragma unroll
  for (int c = 0; c < 3; ++c) { a1[c].x = qrow[4*c + kh]; a1[c].y = qrow[4*c + kh + 1]; }

  // ---- B (weight) fragments; lane column n = m
  const int n = m;
  v2f w1c[3], w2c[4], wsc[2], woc[4];
  #pragma unroll
  for (int c = 0; c < 3; ++c) {
    w1c[c].x = W1[(4*c + kh    ) * 16 + n];
    w1c[c].y = W1[(4*c + kh + 1) * 16 + n];
  }
  #pragma unroll
  for (int c = 0; c < 4; ++c) {            // W2 is 16x8: pad columns 8..15 with 0
    w2c[c].x = (n < 8) ? W2[(4*c + kh    ) * 8 + n] : 0.f;
    w2c[c].y = (n < 8) ? W2[(4*c + kh + 1) * 8 + n] : 0.f;
  }
  #pragma unroll
  for (int c = 0; c < 2; ++c) {            // Ws is 8x16: K=8 -> 2 chunks
    wsc[c].x = Ws[(4*c + kh    ) * 16 + n];
    wsc[c].y = Ws[(4*c + kh + 1) * 16 + n];
  }
  #pragma unroll
  for (int c = 0; c < 4; ++c) {            // Wo is 16x3: pad columns 3..15 with 0
    woc[c].x = (n < 3) ? Wo[(4*c + kh    ) * 3 + n] : 0.f;
    woc[c].y = (n < 3) ? Wo[(4*c + kh + 1) * 3 + n] : 0.f;
  }
  const float b1n = b1[n];
  const float b2n = (n < 8) ? b2[n] : 0.f;
  const float bsn = bs[n];
  const float bon = (n < 3) ? bo[n] : 0.f;

  // ---- layer 1: h = relu(q @ W1 + b1)
  v8f c1 = {};
  c1 = wmma4(a1[0], w1c[0], c1);
  c1 = wmma4(a1[1], w1c[1], c1);
  c1 = wmma4(a1[2], w1c[2], c1);
  #pragma unroll
  for (int v = 0; v < 8; ++v) c1[v] = fmaxf(c1[v] + b1n, 0.f);

  // C-layout -> LDS (row-major tile) -> A-layout fragments
  #pragma unroll
  for (int v = 0; v < 8; ++v) t[(ch + v) * 16 + n] = c1[v];
  asm volatile("s_wait_dscnt 0" ::: "memory");
  v2f a2[4];
  #pragma unroll
  for (int c = 0; c < 4; ++c) { a2[c].x = t[m*16 + 4*c + kh]; a2[c].y = t[m*16 + 4*c + kh + 1]; }

  // ---- layer 2: blk = relu(h @ W2 + b2)   (columns 8..15 stay exactly 0)
  v8f c2 = {};
  #pragma unroll
  for (int c = 0; c < 4; ++c) c2 = wmma4(a2[c], w2c[c], c2);
  #pragma unroll
  for (int v = 0; v < 8; ++v) c2[v] = fmaxf(c2[v] + b2n, 0.f);

  #pragma unroll
  for (int v = 0; v < 8; ++v) t[(ch + v) * 16 + n] = c2[v];
  asm volatile("s_wait_dscnt 0" ::: "memory");
  v2f a3[2];                                // only K=0..7 needed (blk has 8 cols)
  #pragma unroll
  for (int c = 0; c < 2; ++c) { a3[c].x = t[m*16 + 4*c + kh]; a3[c].y = t[m*16 + 4*c + kh + 1]; }

  // ---- layer 3: state = blk @ Ws + bs    (no relu)
  v8f c3 = {};
  c3 = wmma4(a3[0], wsc[0], c3);
  c3 = wmma4(a3[1], wsc[1], c3);
  #pragma unroll
  for (int v = 0; v < 8; ++v) c3[v] += bsn;

  #pragma unroll
  for (int v = 0; v < 8; ++v) t[(ch + v) * 16 + n] = c3[v];
  asm volatile("s_wait_dscnt 0" ::: "memory");
  v2f a4[4];
  #pragma unroll
  for (int c = 0; c < 4; ++c) { a4[c].x = t[m*16 + 4*c + kh]; a4[c].y = t[m*16 + 4*c + kh + 1]; }

  // ---- layer 4: out = state @ Wo + bo    (only N=0..2 valid)
  v8f c4 = {};
  #pragma unroll
  for (int c = 0; c < 4; ++c) c4 = wmma4(a4[c], woc[c], c4);

  if (n < 3) {
    #pragma unroll
    for (int v = 0; v < 8; ++v)
      out[(size_t)(r0 + ch + v) * 3 + n] = c4[v] + bon;
  }

  // ---- force_prev = x[:, 0, 6:9] : 16 rows x 3 comps = 48 values per tile
  for (int i = lane; i < 48; i += 32) {
    const int row = i / 3, comp = i % 3;
    fprev[(size_t)(r0 + row) * 3 + comp] =
        x[(size_t)(r0 + row) * ROW_STRIDE + 6 + comp];
  }
}

extern "C" void kernel_launch(void* const* d_in, const int* in_sizes, int n_in,
                              void* d_out, int out_size, void* d_ws, size_t ws_size,
                              hipStream_t stream) {
  const float* x  = (const float*)d_in[0];
  const float* W1 = (const float*)d_in[1];
  const float* b1 = (const float*)d_in[2];
  const float* W2 = (const float*)d_in[3];
  const float* b2 = (const float*)d_in[4];
  const float* Ws = (const float*)d_in[5];
  const float* bs = (const float*)d_in[6];
  const float* Wo = (const float*)d_in[7];
  const float* bo = (const float*)d_in[8];

  const int B = in_sizes[0] / ROW_STRIDE;          // 262144
  float* out   = (float*)d_out;                    // (B,3)
  float* fprev = out + (size_t)B * 3;              // (B,3) concatenated after

  const int tiles  = (B + 15) / 16;                // 16-row tiles, one per wave
  const int blocks = (tiles + 7) / 8;              // 8 waves (256 thr) per block

  LIMADNN2_mlp_wmma<<<blocks, 256, 0, stream>>>(
      x, W1, b1, W2, b2, Ws, bs, Wo, bo, out, fprev, B);
}